// pen_loss_43542378447221
// MI455X (gfx1250) — compile-verified
//
#include <hip/hip_runtime.h>

#define T 24
#define ML_CONST 10.0f
#define A_PEN_CONST 10.0f

typedef __attribute__((ext_vector_type(2))) float v2f;
typedef __attribute__((ext_vector_type(8))) float v8f;

// Exact f32 wave(32)-sum using V_WMMA_F32_16X16X4_F32.
// A is 16x4: VGPR0 holds A[m][0] (lanes 0-15) and A[m][2] (lanes 16-31);
// VGPR1 holds K=1/K=3 which we zero. B = ones(4x16).
// D[m][n] = A[m][0] + A[m][2] = x[m] + x[m+16]  (same for every n).
// Lane sum of the 8 D VGPRs gives the half-wave sum; shfl_xor(16) completes it.
// EXEC must be all 1s at the call site (full wave, no divergence).
__device__ __forceinline__ float wave_sum_wmma(float x) {
    v2f a;
    a.x = x;
    a.y = 0.0f;
    v2f b;
    b.x = 1.0f;
    b.y = 1.0f;
    v8f c = {};
    c = __builtin_amdgcn_wmma_f32_16x16x4_f32(
        /*neg_a=*/false, a, /*neg_b=*/false, b,
        /*c_mod=*/(short)0, c, /*reuse_a=*/false, /*reuse_b=*/false);
    float s = ((c[0] + c[1]) + (c[2] + c[3])) + ((c[4] + c[5]) + (c[6] + c[7]));
    s += __shfl_xor(s, 16, 32);   // combine the two half-wave sums
    return s;                      // every lane holds the full 32-lane sum
}

// One row per thread: serial cumsum over T=24, hinge penalties, bill sum.
// Rows are contiguous: out row = 96 B, var_y row = 192 B, both 16B-aligned,
// so everything moves as global_load_b128.
__global__ void __launch_bounds__(256)
pen_loss_main(const float* __restrict__ g_out,
              const float* __restrict__ g_vary,
              float* __restrict__ g_partials) {
    const int row = blockIdx.x * blockDim.x + threadIdx.x;
    const float4* dp = reinterpret_cast<const float4*>(g_out)  + (size_t)row * (T / 4);
    const float4* vp = reinterpret_cast<const float4*>(g_vary) + (size_t)row * (2 * T / 4);

    float x0  = 0.0f;   // running cumsum (x0_prev at step t)
    float acc = 0.0f;   // sum of bill over t for this row

#pragma unroll
    for (int c = 0; c < T / 4; ++c) {
        const float4 d4 = dp[c];            // delta[4t..4t+3]
        const float4 p4 = vp[c];            // p
        const float4 a4 = vp[c + T / 4];    // a
        const float dv[4] = {d4.x, d4.y, d4.z, d4.w};
        const float pv[4] = {p4.x, p4.y, p4.z, p4.w};
        const float av[4] = {a4.x, a4.y, a4.z, a4.w};
#pragma unroll
        for (int j = 0; j < 4; ++j) {
            const float d = dv[j];
            float pen = fmaxf(-d - x0, 0.0f)                 // max_v1: hinge(-d - x0_prev)
                      + fmaxf(x0 + d - ML_CONST, 0.0f);      // min_v1: hinge(x0_prev + d - ML)
            if (c != 0 || j != 0)                            // t_mask: t >= 1 (compile-time)
                pen += fmaxf(-d - av[j], 0.0f);              // min_v0: hinge(-d - a)
            // bill = |a + d| * p + pen * A_PEN
            acc = fmaf(fabsf(av[j] + d), pv[j], fmaf(pen, A_PEN_CONST, acc));
            x0 += d;
        }
    }

    // Deterministic block reduction: WMMA wave sum -> LDS -> fixed-order add.
    const float wsum = wave_sum_wmma(acc);

    __shared__ float smem[8];
    const int lane = threadIdx.x & 31;
    const int wid  = threadIdx.x >> 5;
    if (lane == 0) smem[wid] = wsum;
    __syncthreads();
    if (threadIdx.x == 0) {
        float b = 0.0f;
#pragma unroll
        for (int w = 0; w < 8; ++w) b += smem[w];
        g_partials[blockIdx.x] = b;
    }
}

// Single wave (32 threads): fixed-order strided sums of the block partials,
// then one WMMA wave-sum, scale by 1/B. Fully deterministic.
__global__ void __launch_bounds__(32)
pen_loss_finish(const float* __restrict__ g_partials,
                float* __restrict__ g_result,
                int n, float scale) {
    float s = 0.0f;
    for (int i = threadIdx.x; i < n; i += 32)   // n is a multiple of 32
        s += g_partials[i];
    const float tot = wave_sum_wmma(s);
    if (threadIdx.x == 0) g_result[0] = tot * scale;
}

extern "C" void kernel_launch(void* const* d_in, const int* in_sizes, int n_in,
                              void* d_out, int out_size, void* d_ws, size_t ws_size,
                              hipStream_t stream) {
    const float* g_out  = (const float*)d_in[0];   // (B, T)  f32
    const float* g_vary = (const float*)d_in[1];   // (B, 2T) f32
    float* ws  = (float*)d_ws;
    float* res = (float*)d_out;

    const int B = in_sizes[0] / T;        // 1 << 20
    const int threads = 256;
    const int nblocks = B / threads;      // 4096 (B is a power of two)

    pen_loss_main<<<nblocks, threads, 0, stream>>>(g_out, g_vary, ws);
    pen_loss_finish<<<1, 32, 0, stream>>>(ws, res, nblocks, 1.0f / (float)B);
}